// RaggedGravNet_36094905155951
// MI455X (gfx1250) — compile-verified
//
#include <hip/hip_runtime.h>
#include <hip/hip_bf16.h>
#include <math.h>

// ---------------------------------------------------------------------------
// RaggedGravNet for MI455X (gfx1250, wave32).
//  k1: coords = x@W_sp (scalar, tiny) ; feat = relu(x@W_feat+b)  [WMMA f32 16x16x4]
//  k2: per-segment fused distance + top-65 selection.
//      - Gram tile 16x16 with K=4 == exactly one V_WMMA_F32_16X16X4_F32.
//      - Candidate chunks double-buffered in LDS via GLOBAL_LOAD_ASYNC_TO_LDS
//        (ASYNCcnt), overlapping HBM/L2 latency with WMMA + selection.
//      - Per-row top-K kept as a 65-entry max-heap in LDS (O(log K) insert).
//  k3: gather-accumulate (exp weights, mean+max) into LDS f-tile (16x192),
//      then fused output GEMM f@W_out [WMMA f32 16x16x4, K=192].
// ---------------------------------------------------------------------------

typedef __attribute__((ext_vector_type(2))) float v2f;
typedef __attribute__((ext_vector_type(8))) float v8f;

#define N_TOT  32768
#define SEG    8192
#define K_NN   65
#define F_IN   64
#define N_DIM  4
#define N_PROP 64
#define N_FILT 128
#define F_CAT  192   // 2*N_PROP + F_IN

static __device__ __forceinline__ v8f wmma4(v2f a, v2f b, v8f c) {
  // D(16x16,f32) = A(16x4,f32) x B(4x16,f32) + C
  return __builtin_amdgcn_wmma_f32_16x16x4_f32(false, a, false, b,
                                               (short)0, c, false, false);
}

// ===========================================================================
// Kernel 1: coords + squared norms + feat.  128 threads (4 waves) / 16 rows.
// ===========================================================================
__global__ void k1_coords_feat(const float* __restrict__ x,
                               const float* __restrict__ Wsp,
                               const float* __restrict__ Wf,
                               const float* __restrict__ bf,
                               float* __restrict__ coordsOut,
                               float* __restrict__ featWs,
                               float* __restrict__ sqWs) {
  __shared__ float cbuf[16][N_DIM];
  const int base = blockIdx.x * 16;
  const int tid  = threadIdx.x;
  const int wave = tid >> 5;
  const int lane = tid & 31;
  const int half = lane >> 4;
  const int l16  = lane & 15;

  // ---- feat tile: wave w computes cols [16w,16w+16), K=64 in 16 WMMA steps
  const int colBase = wave * 16;
  v8f acc = {};
#pragma unroll
  for (int k0 = 0; k0 < F_IN; k0 += 4) {
    v2f a, b;
    const int row = base + l16;
    a[0] = x[(size_t)row * F_IN + k0 + 2 * half];
    a[1] = x[(size_t)row * F_IN + k0 + 2 * half + 1];
    b[0] = Wf[(size_t)(k0 + 2 * half) * N_PROP + colBase + l16];
    b[1] = Wf[(size_t)(k0 + 2 * half + 1) * N_PROP + colBase + l16];
    acc = wmma4(a, b, acc);
  }
  const float bias = bf[colBase + l16];
#pragma unroll
  for (int r = 0; r < 8; ++r) {
    const int row = base + r + 8 * half;
    featWs[(size_t)row * N_PROP + colBase + l16] = fmaxf(acc[r] + bias, 0.f);
  }

  // ---- coords (N_DIM=4 output cols -> scalar dot products, trivial FLOPs)
  if (tid < 64) {
    const int rr = tid >> 2, dd = tid & 3;
    const int row = base + rr;
    float s = 0.f;
#pragma unroll
    for (int i = 0; i < F_IN; ++i)
      s += x[(size_t)row * F_IN + i] * Wsp[i * N_DIM + dd];
    coordsOut[(size_t)row * N_DIM + dd] = s;
    cbuf[rr][dd] = s;
  }
  __syncthreads();
  if (tid < 16) {
    float s = 0.f;
#pragma unroll
    for (int d = 0; d < N_DIM; ++d) { const float v = cbuf[tid][d]; s += v * v; }
    sqWs[base + tid] = s;
  }
}

// ===========================================================================
// Kernel 2: fused distance + top-K.  128 threads (4 waves), 64 rows per WG.
// ===========================================================================
#define ROWS_WG 64
#define TC      128   // candidate chunk staged in LDS (double-buffered)

__global__ void k2_knn(const float* __restrict__ coords,
                       const float* __restrict__ sq,
                       int* __restrict__ idxOut,
                       float* __restrict__ distOut) {
  __align__(16) __shared__ float candC[2][TC * 4];
  __shared__ float candSq[2][TC];
  __shared__ float d2t[4 * 16 * 17];          // per-wave 16x16 tile, pad 17
  __shared__ float topD[ROWS_WG * K_NN];
  __shared__ int   topI[ROWS_WG * K_NN];

  const int tid  = threadIdx.x;
  const int wave = tid >> 5;
  const int lane = tid & 31;
  const int half = lane >> 4;
  const int l16  = lane & 15;

  const int blocksPerSeg = SEG / ROWS_WG;
  const int seg          = blockIdx.x / blocksPerSeg;
  const int rowBaseLocal = (blockIdx.x % blocksPerSeg) * ROWS_WG;
  const int segBase      = seg * SEG;
  const int waveRowLocal = rowBaseLocal + wave * 16;
  const int waveRowGlob  = segBase + waveRowLocal;

  // A fragment: 16 rows x 4 coords (lanes 0-15: K=0,1 ; lanes 16-31: K=2,3)
  v2f afrag;
  {
    const size_t row = (size_t)(waveRowGlob + l16);
    afrag[0] = coords[row * 4 + 2 * half];
    afrag[1] = coords[row * 4 + 2 * half + 1];
  }
  float rsq[8];
#pragma unroll
  for (int r = 0; r < 8; ++r) rsq[r] = sq[waveRowGlob + r + 8 * half];

  // per-row max-heap of size K_NN in LDS (root = current worst kept dist)
  float* myD = topD + (wave * 16 + l16) * K_NN;
  int*   myI = topI + (wave * 16 + l16) * K_NN;
  float kmax = 3.4e38f;
  if (lane < 16) {
    for (int j = 0; j < K_NN; ++j) { myD[j] = 3.4e38f; myI[j] = segBase; }
  }

  float* waveD2 = d2t + wave * 16 * 17;

  // ---- prologue: async-stage chunk 0 into buffer 0 (ASYNCcnt-tracked)
  {
    const float* gc = coords + (size_t)(segBase + tid) * 4;
    const float* gs = sq + segBase + tid;
    const unsigned lc = (unsigned)(size_t)&candC[0][tid * 4];
    const unsigned ls = (unsigned)(size_t)&candSq[0][tid];
    asm volatile("global_load_async_to_lds_b128 %0, %1, off"
                 :: "v"(lc), "v"(gc) : "memory");
    asm volatile("global_load_async_to_lds_b32 %0, %1, off"
                 :: "v"(ls), "v"(gs) : "memory");
  }

  int ib = 0;
  for (int chunk = 0; chunk < SEG; chunk += TC, ib ^= 1) {
    if (chunk + TC < SEG) {
      // issue next chunk into the other buffer, then wait only for the
      // older pair (async loads complete in order -> asynccnt <= 2)
      const int c = chunk + TC + tid;
      const float* gc = coords + (size_t)(segBase + c) * 4;
      const float* gs = sq + segBase + c;
      const unsigned lc = (unsigned)(size_t)&candC[ib ^ 1][tid * 4];
      const unsigned ls = (unsigned)(size_t)&candSq[ib ^ 1][tid];
      asm volatile("global_load_async_to_lds_b128 %0, %1, off"
                   :: "v"(lc), "v"(gc) : "memory");
      asm volatile("global_load_async_to_lds_b32 %0, %1, off"
                   :: "v"(ls), "v"(gs) : "memory");
      asm volatile("s_wait_asynccnt 0x2" ::: "memory");
    } else {
      asm volatile("s_wait_asynccnt 0x0" ::: "memory");
    }
    __syncthreads();
    const float* curC = candC[ib];
    const float* curS = candSq[ib];

#pragma unroll 1
    for (int sub = 0; sub < TC / 16; ++sub) {
      const int candLocalBase = chunk + sub * 16;
      v2f b;
      b[0] = curC[(sub * 16 + l16) * 4 + 2 * half];
      b[1] = curC[(sub * 16 + l16) * 4 + 2 * half + 1];
      v8f g = {};
      g = wmma4(afrag, b, g);                 // 16x16 Gram tile, one WMMA
      const float csq     = curS[sub * 16 + l16];
      const int   candLoc = candLocalBase + l16;
#pragma unroll
      for (int r = 0; r < 8; ++r) {
        const int rowLocal = waveRowLocal + r + 8 * half;
        float d = rsq[r] + csq - 2.f * g[r];
        d = fmaxf(d, 0.f);
        if (rowLocal == candLoc) d = -1.f;    // self -> always first
        waveD2[(r + 8 * half) * 17 + l16] = d;
      }
      // row owners (lanes 0-15): threshold test + heap insert (O(log K))
      if (lane < 16) {
#pragma unroll 1
        for (int c = 0; c < 16; ++c) {
          const float d = waveD2[l16 * 17 + c];
          if (d < kmax) {
            myD[0] = d;
            myI[0] = segBase + candLocalBase + c;
            int i = 0;
            while (true) {
              const int lft = 2 * i + 1, rgt = 2 * i + 2;
              int big = i;
              float bv = myD[i];
              if (lft < K_NN) { const float v = myD[lft]; if (v > bv) { big = lft; bv = v; } }
              if (rgt < K_NN) { const float v = myD[rgt]; if (v > bv) { big = rgt; bv = v; } }
              if (big == i) break;
              const float td = myD[i]; const int ti = myI[i];
              myD[i] = myD[big]; myI[i] = myI[big];
              myD[big] = td;     myI[big] = ti;
              i = big;
            }
            kmax = myD[0];
          }
        }
      }
    }
    __syncthreads();
  }

  // selection-sort ascending, emit idx/dist (dist clamped to >= 0)
  if (lane < 16) {
    const size_t row = (size_t)(waveRowGlob + l16);
    for (int j = 0; j < K_NN; ++j) {
      float m = 3.5e38f; int mp = j;
      for (int t = j; t < K_NN; ++t) {
        const float v = myD[t];
        if (v < m) { m = v; mp = t; }
      }
      const float dj = myD[j]; const int ij = myI[j];
      myD[j] = myD[mp]; myI[j] = myI[mp];
      myD[mp] = dj;     myI[mp] = ij;
      idxOut[row * K_NN + j]  = myI[j];
      distOut[row * K_NN + j] = fmaxf(myD[j], 0.f);
    }
  }
}

// ===========================================================================
// Kernel 3: gather/accumulate + fused output GEMM.  256 threads, 16 rows/WG.
// ===========================================================================
#define FSTR 196   // padded LDS row stride (196 % 64 != 0 -> conflict-free)

__global__ void k3_accum_out(const float* __restrict__ x,
                             const float* __restrict__ featWs,
                             const int* __restrict__ idxOut,
                             const float* __restrict__ distOut,
                             const float* __restrict__ Wout,
                             const float* __restrict__ bout,
                             float* __restrict__ outF) {
  __shared__ float fbuf[16 * FSTR];
  const int tid  = threadIdx.x;
  const int base = blockIdx.x * 16;

  // ---- Phase A: 16 threads/node, 4 feat dims per thread
  {
    const int node = tid >> 4;
    const int t16  = tid & 15;
    const int row  = base + node;
    const int d0   = t16 * 4;
    float am0 = 0, am1 = 0, am2 = 0, am3 = 0;
    float ax0 = 0, ax1 = 0, ax2 = 0, ax3 = 0;
    const int*   ip = idxOut  + (size_t)row * K_NN;
    const float* dp = distOut + (size_t)row * K_NN;
#pragma unroll 1
    for (int j = 1; j < K_NN; ++j) {          // skip col 0 (self)
      const int   g = ip[j];
      const float w = __expf(-10.f * dp[j]);
      const float4 f4 = *(const float4*)(featWs + (size_t)g * N_PROP + d0);
      am0 += w * f4.x; am1 += w * f4.y; am2 += w * f4.z; am3 += w * f4.w;
      ax0 = fmaxf(ax0, w * f4.x); ax1 = fmaxf(ax1, w * f4.y);
      ax2 = fmaxf(ax2, w * f4.z); ax3 = fmaxf(ax3, w * f4.w);
    }
    const float4 fs = *(const float4*)(featWs + (size_t)row * N_PROP + d0);
    const float4 xs = *(const float4*)(x      + (size_t)row * F_IN  + d0);
    const float inv = 1.f / (float)(K_NN - 1);
    float* fb = fbuf + node * FSTR;
    fb[d0 + 0]       = am0 * inv - fs.x;  fb[d0 + 1]       = am1 * inv - fs.y;
    fb[d0 + 2]       = am2 * inv - fs.z;  fb[d0 + 3]       = am3 * inv - fs.w;
    fb[64 + d0 + 0]  = ax0 - fs.x;        fb[64 + d0 + 1]  = ax1 - fs.y;
    fb[64 + d0 + 2]  = ax2 - fs.z;        fb[64 + d0 + 3]  = ax3 - fs.w;
    fb[128 + d0 + 0] = xs.x;              fb[128 + d0 + 1] = xs.y;
    fb[128 + d0 + 2] = xs.z;              fb[128 + d0 + 3] = xs.w;
  }
  __syncthreads();

  // ---- Phase B: wave w -> out cols [16w,16w+16), K=192 in 48 WMMA steps
  const int wave = tid >> 5;
  const int lane = tid & 31;
  const int half = lane >> 4;
  const int l16  = lane & 15;
  const int colBase = wave * 16;
  v8f acc = {};
#pragma unroll 1
  for (int k0 = 0; k0 < F_CAT; k0 += 4) {
    v2f a, b;
    a[0] = fbuf[l16 * FSTR + k0 + 2 * half];
    a[1] = fbuf[l16 * FSTR + k0 + 2 * half + 1];
    b[0] = Wout[(size_t)(k0 + 2 * half) * N_FILT + colBase + l16];
    b[1] = Wout[(size_t)(k0 + 2 * half + 1) * N_FILT + colBase + l16];
    acc = wmma4(a, b, acc);
  }
  const float bias = bout[colBase + l16];
#pragma unroll
  for (int r = 0; r < 8; ++r) {
    const int row = base + r + 8 * half;
    outF[(size_t)row * N_FILT + colBase + l16] = fmaxf(acc[r] + bias, 0.f);
  }
}

// ===========================================================================
extern "C" void kernel_launch(void* const* d_in, const int* in_sizes, int n_in,
                              void* d_out, int out_size, void* d_ws, size_t ws_size,
                              hipStream_t stream) {
  // setup_inputs order: x, row_splits, W_feat, b_feat, W_sp, W_out, b_out
  const float* x    = (const float*)d_in[0];
  const float* Wf   = (const float*)d_in[2];
  const float* bf   = (const float*)d_in[3];
  const float* Wsp  = (const float*)d_in[4];
  const float* Wout = (const float*)d_in[5];
  const float* bout = (const float*)d_in[6];

  // outputs concatenated flat: outfeats, coords, idx(int32 bits), dist
  float* outF      = (float*)d_out;
  float* coordsOut = outF + (size_t)N_TOT * N_FILT;
  int*   idxOut    = (int*)(coordsOut + (size_t)N_TOT * N_DIM);
  float* distOut   = (float*)(idxOut + (size_t)N_TOT * K_NN);

  // workspace: feat (N x 64 f32) + sq (N f32)
  float* featWs = (float*)d_ws;
  float* sqWs   = featWs + (size_t)N_TOT * N_PROP;

  k1_coords_feat<<<N_TOT / 16, 128, 0, stream>>>(x, Wsp, Wf, bf,
                                                 coordsOut, featWs, sqWs);
  k2_knn<<<N_TOT / ROWS_WG, 128, 0, stream>>>(coordsOut, sqWs, idxOut, distOut);
  k3_accum_out<<<N_TOT / 16, 256, 0, stream>>>(x, featWs, idxOut, distOut,
                                               Wout, bout, outF);
}